// MaxMargin_Loss_9259949491057
// MI455X (gfx1250) — compile-verified
//
#include <hip/hip_runtime.h>
#include <math.h>

#define SIM_NEG 25
#define BATCH   8192
#define DIM     512
#define MARGIN  1.0f
#define EPS     1e-8f

// LDS row stride in floats: 516 mod 64 == 4 -> the 16 lanes of each half-wave
// hit disjoint bank pairs for b64 reads (all 64 banks used exactly once).
#define LDS_STRIDE 516

typedef __attribute__((ext_vector_type(2))) float v2f;
typedef __attribute__((ext_vector_type(8))) float v8f;

// ---------------------------------------------------------------------------
// Gram accumulate: D(16x16) += A(16x512) * B(512x16) via 128 chained
// v_wmma_f32_16x16x4_f32 ops. a2/b2 point at this lane's float2 slot:
//   element(lane, v) = M[row(lane%16)][kb + v + 2*(lane>=16)]
// which is the identical mapping for both the A and B operands of the x4 f32
// WMMA shape (ISA 7.12.2), so one addressing function serves both.
// ---------------------------------------------------------------------------
__device__ __forceinline__ v8f gram512(const float2* a2, const float2* b2) {
    v8f c = {};
#pragma unroll 8
    for (int k = 0; k < DIM / 2; k += 2) {   // k in float2 units; 4 floats/step
        float2 av = a2[k];
        float2 bv = b2[k];
        v2f a; a[0] = av.x; a[1] = av.y;
        v2f b; b[0] = bv.x; b[1] = bv.y;
        // 8 args: (neg_a, A, neg_b, B, c_mod, C, reuse_a, reuse_b)
        c = __builtin_amdgcn_wmma_f32_16x16x4_f32(
                false, a, false, b, (short)0, c, false, false);
    }
    return c;
}

// Diagonal of the 16x16 f32 C/D tile: row m -> VGPR (m&7) of lane m (m<8)
// or lane m+16 (m>=8). Lanes 0-7 get rows 0-7, lanes 24-31 get rows 8-15.
__device__ __forceinline__ float diag_of(const v8f& c, int lane) {
    float d = 0.0f;
#pragma unroll
    for (int i = 0; i < 8; ++i) {
        if (lane == i)      d = c[i];
        if (lane == 24 + i) d = c[i];
    }
    return d;
}

// ---------------------------------------------------------------------------
// Kernel 1: per-row L2 norms of y_pred and y_true. One wave per row.
// ---------------------------------------------------------------------------
__global__ __launch_bounds__(256)
void norms_kernel(const float* __restrict__ y_pred,
                  const float* __restrict__ y_true,
                  float* __restrict__ n_pred,
                  float* __restrict__ n_true) {
    const int wave = threadIdx.x >> 5;
    const int lane = threadIdx.x & 31;
    const int row  = blockIdx.x * 8 + wave;

    const float4* p = (const float4*)(y_pred + (size_t)row * DIM);
    const float4* t = (const float4*)(y_true + (size_t)row * DIM);
    float sp = 0.0f, st = 0.0f;
#pragma unroll
    for (int i = lane; i < DIM / 4; i += 32) {
        float4 a = p[i];
        float4 b = t[i];
        sp += a.x * a.x + a.y * a.y + a.z * a.z + a.w * a.w;
        st += b.x * b.x + b.y * b.y + b.z * b.z + b.w * b.w;
    }
#pragma unroll
    for (int off = 16; off; off >>= 1) {
        sp += __shfl_xor(sp, off);
        st += __shfl_xor(st, off);
    }
    if (lane == 0) {
        n_pred[row] = sqrtf(sp);
        n_true[row] = sqrtf(st);
    }
}

// ---------------------------------------------------------------------------
// Kernel 2: one block (4 waves) per 16-row batch tile. Pred tile staged in
// LDS once; each wave runs the positive Gram plus a strided subset of the 25
// negative-sampling rounds. Writes one deterministic partial sum per block.
// ---------------------------------------------------------------------------
__global__ __launch_bounds__(128)
void margin_kernel(const float* __restrict__ y_pred,
                   const float* __restrict__ y_true,
                   const int*   __restrict__ perm,
                   const float* __restrict__ n_pred,
                   const float* __restrict__ n_true,
                   float* __restrict__ partials) {
    __shared__ float predA[16 * LDS_STRIDE];
    __shared__ float wsum[4];

    const int row0 = blockIdx.x * 16;
    const int tid  = threadIdx.x;     // 0..127
    const int wave = tid >> 5;        // 0..3
    const int lane = tid & 31;
    const int col  = lane & 15;       // matrix row/col index for this lane
    const int half = lane >> 4;       // which K-pair half

    // --- cooperative stage of the 16x512 pred tile into padded LDS ---
    {
        const float4* src = (const float4*)(y_pred + (size_t)row0 * DIM);
        for (int i = tid; i < 16 * DIM / 4; i += 128) {
            const int r = i / (DIM / 4);
            const int c = i % (DIM / 4);
            float4 v = src[i];
            float* dst = &predA[r * LDS_STRIDE + c * 4];
            dst[0] = v.x; dst[1] = v.y; dst[2] = v.z; dst[3] = v.w;
        }
    }
    __syncthreads();

    // This lane's A-operand slot (LDS) -- shared by every Gram round.
    const float2* a2 = (const float2*)(&predA[col * LDS_STRIDE + 2 * half]);

    const bool have = (lane < 8) || (lane >= 24);       // owns a diag element
    const int  m    = (lane < 8) ? lane : (lane - 16);  // its batch-tile row

    // --- positive round: diag(Pred * True^T) on the aligned tile ---
    const float2* bpos =
        (const float2*)(y_true + (size_t)(row0 + col) * DIM + 2 * half);
    v8f cpos = gram512(a2, bpos);
    float dpos = diag_of(cpos, lane);

    float npd = 0.0f, cos_pos = 0.0f;
    if (have) {
        npd = n_pred[row0 + m];
        const float ntr = n_true[row0 + m];
        cos_pos = dpos / fmaxf(ntr * npd, EPS);
    }

    // --- negative rounds split across the 4 waves (uniform per wave) ---
    float cost = 0.0f;
    for (int s = wave; s < SIM_NEG; s += 4) {
        const int rowB = perm[(size_t)s * BATCH + row0 + col];
        const float2* bneg =
            (const float2*)(y_true + (size_t)rowB * DIM + 2 * half);
        v8f cneg = gram512(a2, bneg);
        float dneg = diag_of(cneg, lane);
        if (have) {
            // diag-owning lane's col == its row m, so rowB == perm[s, row0+m]
            const float ntn = n_true[rowB];
            const float cos_neg = dneg / fmaxf(ntn * npd, EPS);
            cost += fmaxf(MARGIN - cos_pos + cos_neg, 0.0f);
        }
    }

    // --- deterministic block reduction (no float atomics) ---
#pragma unroll
    for (int off = 16; off; off >>= 1) cost += __shfl_xor(cost, off);
    if (lane == 0) wsum[wave] = cost;
    __syncthreads();
    if (tid == 0)
        partials[blockIdx.x] = wsum[0] + wsum[1] + wsum[2] + wsum[3];
}

// ---------------------------------------------------------------------------
// Kernel 3: fixed-order reduction of the 512 block partials -> scalar loss.
// ---------------------------------------------------------------------------
__global__ __launch_bounds__(256)
void finalize_kernel(const float* __restrict__ partials,
                     float* __restrict__ out) {
    __shared__ float red[256];
    const int t = threadIdx.x;
    red[t] = partials[t] + partials[t + 256];
    __syncthreads();
#pragma unroll
    for (int off = 128; off; off >>= 1) {
        if (t < off) red[t] += red[t + off];
        __syncthreads();
    }
    if (t == 0) out[0] = red[0] / (float)(BATCH * SIM_NEG);
}

// ---------------------------------------------------------------------------
extern "C" void kernel_launch(void* const* d_in, const int* in_sizes, int n_in,
                              void* d_out, int out_size, void* d_ws, size_t ws_size,
                              hipStream_t stream) {
    const float* y_pred = (const float*)d_in[0];
    const float* y_true = (const float*)d_in[1];
    const int*   perm   = (const int*)d_in[2];

    float* ws       = (float*)d_ws;
    float* partials = ws;                       // 512 floats
    float* n_pred   = ws + 512;                 // 8192 floats
    float* n_true   = ws + 512 + BATCH;         // 8192 floats

    norms_kernel   <<<BATCH / 8, 256, 0, stream>>>(y_pred, y_true, n_pred, n_true);
    margin_kernel  <<<BATCH / 16, 128, 0, stream>>>(y_pred, y_true, perm,
                                                    n_pred, n_true, partials);
    finalize_kernel<<<1, 256, 0, stream>>>(partials, (float*)d_out);
}